// decoder_34299608826608
// MI455X (gfx1250) — compile-verified
//
#include <hip/hip_runtime.h>
#include <hip/hip_bf16.h>

typedef __attribute__((ext_vector_type(2))) float v2f;
typedef __attribute__((ext_vector_type(8))) float v8f;

// Problem constants
#define BB 128
#define SS 2048
#define HH 512
#define OO 32000

// GEMM tiling: 256 threads = 8 waves (2x4); each wave owns a 16x32 C tile
// (two 16x16 WMMA accumulators sharing one A fragment).
#define BM 32
#define BN 128
#define KC 32
#define LDSS 36   // LDS row stride (floats): 16B-aligned rows, conflict-free frag reads

// Async global->LDS copy of one float4, no VGPR data registers (CDNA5 ASYNCcnt path)
__device__ __forceinline__ void async_cp_b128(void* lds_dst, const void* gsrc) {
    unsigned ldsoff = (unsigned)(unsigned long long)lds_dst;  // low 32 bits = LDS offset
    asm volatile("global_load_async_to_lds_b128 %0, %1, off"
                 :: "v"(ldsoff), "v"(gsrc) : "memory");
}
__device__ __forceinline__ void async_wait0() {
    asm volatile("s_wait_asynccnt 0x0" ::: "memory");
}

// C[M,N] = A[M,K] @ Bt[N,K]^T + bias[N]   (torch-style  x @ W.T + b)
__global__ __launch_bounds__(256) void gemm_bt_bias(
    const float* __restrict__ A, const float* __restrict__ Bt,
    const float* __restrict__ bias, float* __restrict__ C,
    int M, int N, int K)
{
    __shared__ float sa[2][BM * LDSS];
    __shared__ float sb[2][BN * LDSS];

    const int tid  = threadIdx.x;
    const int wave = tid >> 5;
    const int lane = tid & 31;
    const int wm   = wave >> 2;      // 0..1
    const int wn   = wave & 3;       // 0..3
    const int lh   = lane & 15;
    const int lg   = lane >> 4;

    const int m0 = blockIdx.y * BM;
    const int n0 = blockIdx.x * BN;

    // staging geometry: tile rows are KC=32 floats = 8 float4; 256 threads
    const int ra = tid >> 3;               // 0..31    (A row)
    const int ca = (tid & 7) * 4;          // 0,4..28  (A col)
    const float* Agp = A + (size_t)(m0 + ra) * K + ca;   // + k0

    v8f acc0 = {0.f,0.f,0.f,0.f,0.f,0.f,0.f,0.f};
    v8f acc1 = {0.f,0.f,0.f,0.f,0.f,0.f,0.f,0.f};

    // prologue: async-copy slab 0 into buffer 0
    async_cp_b128(&sa[0][ra * LDSS + ca], Agp);
    #pragma unroll
    for (int i = 0; i < 4; ++i) {
        const int idx = tid + i * 256;
        const int r = idx >> 3, c = (idx & 7) * 4;
        async_cp_b128(&sb[0][r * LDSS + c], Bt + (size_t)(n0 + r) * K + c);
    }
    async_wait0();
    __syncthreads();

    int buf = 0;
    for (int k0 = 0; k0 < K; k0 += KC) {
        const int nk = k0 + KC;
        const bool more = (nk < K);
        if (more) {
            // issue next slab's DMA into the other buffer; in flight during WMMA
            const int nb = buf ^ 1;
            async_cp_b128(&sa[nb][ra * LDSS + ca], Agp + nk);
            #pragma unroll
            for (int i = 0; i < 4; ++i) {
                const int idx = tid + i * 256;
                const int r = idx >> 3, c = (idx & 7) * 4;
                async_cp_b128(&sb[nb][r * LDSS + c],
                              Bt + (size_t)(n0 + r) * K + nk + c);
            }
        }

        const float* pa  = &sa[buf][(wm * 16 + lh) * LDSS + 2 * lg];
        const float* pb0 = &sb[buf][(wn * 32 + lh) * LDSS + 2 * lg];
        const float* pb1 = &sb[buf][(wn * 32 + 16 + lh) * LDSS + 2 * lg];
        #pragma unroll
        for (int kk = 0; kk < KC; kk += 4) {
            v2f a  = *(const v2f*)(pa  + kk);   // ds_load_b64, 8B-aligned
            v2f b0 = *(const v2f*)(pb0 + kk);
            v2f b1 = *(const v2f*)(pb1 + kk);
            acc0 = __builtin_amdgcn_wmma_f32_16x16x4_f32(
                false, a, false, b0, (short)0, acc0, false, false);
            acc1 = __builtin_amdgcn_wmma_f32_16x16x4_f32(
                false, a, false, b1, (short)0, acc1, false, false);
        }

        if (more) async_wait0();   // own wave's DMA done; barrier publishes all waves'
        __syncthreads();
        if (more) buf ^= 1;
    }

    // C/D layout: vgpr r, lane L -> m = r + 8*(L/16), n = L%16
    const int nA = n0 + wn * 32 + lh;
    const int nB = nA + 16;
    const float bvA = bias[nA];
    const float bvB = bias[nB];
    const int mr = m0 + wm * 16 + lg * 8;
    #pragma unroll
    for (int r = 0; r < 8; ++r) {
        C[(size_t)(mr + r) * N + nA] = acc0[r] + bvA;
        C[(size_t)(mr + r) * N + nB] = acc1[r] + bvB;
    }
}

// h = (1-z)*n + z*h_prev  with r,z,n from precomputed gx, gh (PyTorch gate order)
__global__ __launch_bounds__(256) void gru_gate(
    const float* __restrict__ gx, const float* __restrict__ gh,
    const float* __restrict__ hprev, float* __restrict__ hout)
{
    int i = blockIdx.x * blockDim.x + threadIdx.x;   // 0 .. B*H-1
    int b = i >> 9;          // /H
    int j = i & (HH - 1);    // %H
    const float* gxr = gx + (size_t)b * 3 * HH;
    const float* ghr = gh + (size_t)b * 3 * HH;
    float r = 1.f / (1.f + __expf(-(gxr[j] + ghr[j])));
    float z = 1.f / (1.f + __expf(-(gxr[HH + j] + ghr[HH + j])));
    float n = tanhf(gxr[2 * HH + j] + r * ghr[2 * HH + j]);
    hout[i] = (1.f - z) * n + z * hprev[i];
}

// Fused attention: one pass over encoder_output (512 MB read ONCE).
// Block = one batch row, 512 threads = 16 waves; wave w owns s in [w*128,(w+1)*128).
// Online softmax per wave, LDS combine across waves. Writes attn and
// cat = [dec_out, context].
__global__ __launch_bounds__(512) void attn_fused(
    const float* __restrict__ dec, const float* __restrict__ enc,
    const int* __restrict__ mask, float* __restrict__ attn,
    float* __restrict__ cat)
{
    __shared__ float s_m[16];
    __shared__ float s_l[16];
    __shared__ float s_ctx[16 * HH];

    const int b    = blockIdx.x;
    const int tid  = threadIdx.x;
    const int wave = tid >> 5;
    const int lane = tid & 31;

    // dec_out fragment: lane holds h = lane*16 .. lane*16+15
    float dreg[16];
    {
        const float4* dp = (const float4*)(dec + (size_t)b * HH + lane * 16);
        float4 d0 = dp[0], d1 = dp[1], d2 = dp[2], d3 = dp[3];
        dreg[0]=d0.x; dreg[1]=d0.y; dreg[2]=d0.z; dreg[3]=d0.w;
        dreg[4]=d1.x; dreg[5]=d1.y; dreg[6]=d1.z; dreg[7]=d1.w;
        dreg[8]=d2.x; dreg[9]=d2.y; dreg[10]=d2.z; dreg[11]=d2.w;
        dreg[12]=d3.x; dreg[13]=d3.y; dreg[14]=d3.z; dreg[15]=d3.w;
    }

    float m = -__builtin_inff();
    float l = 0.f;
    float cacc[16];
    #pragma unroll
    for (int j = 0; j < 16; ++j) cacc[j] = 0.f;
    float sreg[4];

    const float* encb = enc + (size_t)b * SS * HH;
    const int*   mrow = mask + (size_t)b * SS;

    for (int i = 0; i < 128; ++i) {
        const int s = wave * 128 + i;
        const float4* ep = (const float4*)(encb + (size_t)s * HH + lane * 16);
        float4 e0 = ep[0], e1 = ep[1], e2 = ep[2], e3 = ep[3];
        float er[16] = {e0.x,e0.y,e0.z,e0.w, e1.x,e1.y,e1.z,e1.w,
                        e2.x,e2.y,e2.z,e2.w, e3.x,e3.y,e3.z,e3.w};
        float dot = 0.f;
        #pragma unroll
        for (int j = 0; j < 16; ++j) dot += dreg[j] * er[j];
        // wave32 tree reduction -> all lanes get the full dot
        #pragma unroll
        for (int off = 16; off > 0; off >>= 1)
            dot += __shfl_xor(dot, off, 32);

        float score = (mrow[s] == 1) ? -1000000000.0f : dot;
        if (lane == (i & 31)) sreg[i >> 5] = score;   // keep raw score for attn output

        // online softmax update (p broadcast across lanes)
        float mn = fmaxf(m, score);
        float sc = __expf(m - mn);       // first iter: exp(-inf)=0
        float p  = __expf(score - mn);
        l = l * sc + p;
        #pragma unroll
        for (int j = 0; j < 16; ++j) cacc[j] = cacc[j] * sc + p * er[j];
        m = mn;
    }

    if (lane == 0) { s_m[wave] = m; s_l[wave] = l; }
    #pragma unroll
    for (int j = 0; j < 16; ++j)
        s_ctx[wave * HH + lane * 16 + j] = cacc[j];
    __syncthreads();

    // combine 16 waves (every thread computes identical M, L)
    float M = -__builtin_inff();
    #pragma unroll
    for (int w = 0; w < 16; ++w) M = fmaxf(M, s_m[w]);
    float L = 0.f;
    #pragma unroll
    for (int w = 0; w < 16; ++w) L += __expf(s_m[w] - M) * s_l[w];
    const float Linv = 1.0f / L;

    // context: thread t owns h = t  (512 threads <-> 512 h)
    float ch = 0.f;
    #pragma unroll
    for (int w = 0; w < 16; ++w) ch += __expf(s_m[w] - M) * s_ctx[w * HH + tid];
    cat[(size_t)b * 2 * HH + HH + tid] = ch * Linv;      // cat[:,512:] = context
    cat[(size_t)b * 2 * HH + tid]      = dec[(size_t)b * HH + tid]; // cat[:,:512]

    // attn weights: lane L slot k holds score for s = wave*128 + k*32 + L
    #pragma unroll
    for (int k = 0; k < 4; ++k) {
        int s = wave * 128 + k * 32 + lane;
        attn[(size_t)b * SS + s] = __expf(sreg[k] - M) * Linv;
    }
}

extern "C" void kernel_launch(void* const* d_in, const int* in_sizes, int n_in,
                              void* d_out, int out_size, void* d_ws, size_t ws_size,
                              hipStream_t stream) {
    const float* dec_in   = (const float*)d_in[0];   // (B,1,H)
    const float* enc      = (const float*)d_in[1];   // (B,S,H)
    const float* h_last   = (const float*)d_in[2];   // (L,B,H)
    const int*   mask     = (const int*)  d_in[3];   // (B,S)
    const float* w_ih     = (const float*)d_in[4];   // (L,3H,H)
    const float* w_hh     = (const float*)d_in[5];   // (L,3H,H)
    const float* b_ih     = (const float*)d_in[6];   // (L,3H)
    const float* b_hh     = (const float*)d_in[7];   // (L,3H)
    const float* W_concat = (const float*)d_in[8];   // (H,2H)
    const float* b_concat = (const float*)d_in[9];   // (H)
    const float* W_cls    = (const float*)d_in[10];  // (O,H)
    const float* b_cls    = (const float*)d_in[11];  // (O)

    float* out = (float*)d_out;                      // logits (B*O) then attn (B*S)
    float* ws  = (float*)d_ws;

    float* gx   = ws;                   // B*3H
    float* gh   = gx  + BB * 3 * HH;    // B*3H
    float* x0   = gh  + BB * 3 * HH;    // B*H  (h after layer 0)
    float* x1   = x0  + BB * HH;        // B*H  (h after layer 1)
    float* decb = x1  + BB * HH;        // B*H  (dec_out = h after layer 2)
    float* cat  = decb + BB * HH;       // B*2H
    float* out1 = cat + BB * 2 * HH;    // B*H

    const float* xs[3] = {dec_in, x0, x1};
    float*       hs[3] = {x0, x1, decb};

    const dim3 gGates(3 * HH / BN, BB / BM);   // (12, 4)
    for (int l = 0; l < 3; ++l) {
        const float* hp = h_last + (size_t)l * BB * HH;
        gemm_bt_bias<<<gGates, 256, 0, stream>>>(
            xs[l], w_ih + (size_t)l * 3 * HH * HH, b_ih + (size_t)l * 3 * HH,
            gx, BB, 3 * HH, HH);
        gemm_bt_bias<<<gGates, 256, 0, stream>>>(
            hp, w_hh + (size_t)l * 3 * HH * HH, b_hh + (size_t)l * 3 * HH,
            gh, BB, 3 * HH, HH);
        gru_gate<<<(BB * HH) / 256, 256, 0, stream>>>(gx, gh, hp, hs[l]);
    }

    // fused scores + online softmax + context; writes attn output and cat
    attn_fused<<<BB, 512, 0, stream>>>(decb, enc, mask, out + (size_t)BB * OO, cat);

    // out1 = cat @ W_concat^T + b_concat   (M=128, N=512, K=1024)
    gemm_bt_bias<<<dim3(HH / BN, BB / BM), 256, 0, stream>>>(
        cat, W_concat, b_concat, out1, BB, HH, 2 * HH);

    // logits = out1 @ W_cls^T + b_cls      (M=128, N=32000, K=512)
    gemm_bt_bias<<<dim3(OO / BN, BB / BM), 256, 0, stream>>>(
        out1, W_cls, b_cls, out, BB, OO, HH);
}